// RecursiveFilter_63909113364758
// MI455X (gfx1250) — compile-verified
//
#include <hip/hip_runtime.h>
#include <cstdint>
#include <cstddef>

// ---------------------------------------------------------------------------
// IIR scan y[0]=x[0]; y[t] = w*x[t] + (1-w)*y[t-1] along axis 2 of
// (4, 8, 64, 256, 256) f32.  Memory-bound: 1.07 GB traffic -> ~46 us floor
// at 23.3 TB/s.  Strategy: coalesced float4 streaming with a 4-deep
// global_load_async_to_lds_b128 pipeline (ASYNCcnt), serial IIR in VGPRs,
// nontemporal b128 stores.  All grid factors are compile-time constants so
// the prologue is pure shift/mask (no SALU division).
// ---------------------------------------------------------------------------

typedef __attribute__((ext_vector_type(4))) float v4f;

#define IIR_W   0.8f
#define T_STEPS 64
#define PLANE   65536              // 256*256 spatial elements per time slice
#define BLOCK   256                // 8 wave32s
#define VEC     4                  // floats per thread per step (b128)
#define TILE    (BLOCK * VEC)      // 1024 floats = 4 KiB staged per step
#define DEPTH   4                  // async pipeline depth (ring of 4 tiles)
#define CHUNKS_PER_PLANE (PLANE / TILE)   // 64  -> shift by 6
#define CHUNK_SHIFT 6
#define CHUNK_MASK  (CHUNKS_PER_PLANE - 1)

// CDNA5 async global->LDS copy (ASYNCcnt-tracked).  vdst VGPR carries the
// wave-relative LDS byte address, vaddr VGPR pair carries the 64-bit global
// address (GV mode, saddr = off).
#define ISSUE_ASYNC_B128(ldsoff, gaddr64)                                    \
  asm volatile("global_load_async_to_lds_b128 %0, %1, off"                   \
               :: "v"(ldsoff), "v"(gaddr64) : "memory")

#define WAIT_ASYNC(n) asm volatile("s_wait_asynccnt " #n ::: "memory")
#define WAIT_DS0()    asm volatile("s_wait_dscnt 0" ::: "memory")

__global__ __launch_bounds__(BLOCK)
void iir_scan_async(const float* __restrict__ x, float* __restrict__ y) {
  __shared__ __align__(16) float buf[DEPTH][TILE];

  const int tid   = (int)threadIdx.x;
  const int plane = (int)(blockIdx.x >> CHUNK_SHIFT);      // (bs,c) flattened
  const int chunk = (int)(blockIdx.x & CHUNK_MASK);

  const size_t base = (size_t)plane * (size_t)T_STEPS * (size_t)PLANE
                    + (size_t)chunk * (size_t)TILE
                    + (size_t)tid * (size_t)VEC;
  const float* gx = x + base;
  float*       gy = y + base;

  // LDS byte offset of this lane's 16B slot in ring buffer 0.  Generic
  // pointers to LDS carry the wave-relative LDS offset in addr[31:0].
  const unsigned lds0 = (unsigned)(size_t)(&buf[0][tid * VEC]);

  // ---- prologue: fill the 4-deep pipeline -------------------------------
#pragma unroll
  for (int k = 0; k < DEPTH; ++k) {
    ISSUE_ASYNC_B128(lds0 + (unsigned)(k * TILE * 4),
                     (unsigned long long)(size_t)(gx + (size_t)k * PLANE));
  }

  const float w = IIR_W;
  const float u = 1.0f - IIR_W;
  v4f acc;

  // Streaming pointers: next global tile to fetch, next output slot.
  const float* gnext = gx + (size_t)DEPTH * PLANE;   // tile t+DEPTH source
  float*       gst   = gy;                           // output for step t

  // ---- t = 0 : y[0] = x[0] ----------------------------------------------
  {
    WAIT_ASYNC(3);                                    // tile 0 resident
    acc = *(const v4f*)(&buf[0][tid * VEC]);          // ds_load_b128
    WAIT_DS0();                                       // close WAR on buffer 0
    ISSUE_ASYNC_B128(lds0, (unsigned long long)(size_t)gnext);
    __builtin_nontemporal_store(acc, (v4f*)gst);
    gnext += PLANE;
    gst   += PLANE;
  }

  // ---- main loop: t = 1 .. T_STEPS-DEPTH-1, issue t+DEPTH each step -----
  for (int t = 1; t < T_STEPS - DEPTH; ++t) {
    WAIT_ASYNC(3);                                    // tile t resident
    const v4f xi = *(const v4f*)(&buf[t & (DEPTH - 1)][tid * VEC]);
    WAIT_DS0();                                       // read done before reuse
    ISSUE_ASYNC_B128(lds0 + (unsigned)((t & (DEPTH - 1)) * TILE * 4),
                     (unsigned long long)(size_t)gnext);
    acc = w * xi + u * acc;                           // v_fma chain (contract)
    __builtin_nontemporal_store(acc, (v4f*)gst);
    gnext += PLANE;
    gst   += PLANE;
  }

  // ---- tail: drain pipeline (waits 3,2,1,0), no new issues --------------
  {
    int t = T_STEPS - DEPTH;                          // 60
    WAIT_ASYNC(3);
    v4f xi = *(const v4f*)(&buf[t & (DEPTH - 1)][tid * VEC]);
    acc = w * xi + u * acc;
    __builtin_nontemporal_store(acc, (v4f*)gst);
    gst += PLANE; ++t;                                // 61
    WAIT_ASYNC(2);
    xi = *(const v4f*)(&buf[t & (DEPTH - 1)][tid * VEC]);
    acc = w * xi + u * acc;
    __builtin_nontemporal_store(acc, (v4f*)gst);
    gst += PLANE; ++t;                                // 62
    WAIT_ASYNC(1);
    xi = *(const v4f*)(&buf[t & (DEPTH - 1)][tid * VEC]);
    acc = w * xi + u * acc;
    __builtin_nontemporal_store(acc, (v4f*)gst);
    gst += PLANE; ++t;                                // 63
    WAIT_ASYNC(0);
    xi = *(const v4f*)(&buf[t & (DEPTH - 1)][tid * VEC]);
    acc = w * xi + u * acc;
    __builtin_nontemporal_store(acc, (v4f*)gst);
  }
}

extern "C" void kernel_launch(void* const* d_in, const int* in_sizes, int n_in,
                              void* d_out, int out_size, void* d_ws, size_t ws_size,
                              hipStream_t stream) {
  (void)n_in; (void)out_size; (void)d_ws; (void)ws_size;

  const float* x = (const float*)d_in[0];
  float*       y = (float*)d_out;

  // total = bs*c*T*H*W ; planes = bs*c (= 32 for the reference shape)
  const long long total  = (long long)in_sizes[0];
  const int       planes = (int)(total / ((long long)T_STEPS * PLANE));

  dim3 grid((unsigned)(planes * CHUNKS_PER_PLANE));   // 2048 blocks
  dim3 block(BLOCK);                                  // 8 wave32s
  iir_scan_async<<<grid, block, 0, stream>>>(x, y);
}